// DeepFM_42159398977905
// MI455X (gfx1250) — compile-verified
//
#include <hip/hip_runtime.h>
#include <hip/hip_bf16.h>
#include <math.h>

// ---------------------------------------------------------------------------
// DeepFM forward, MI455X (gfx1250, wave32).
// Bandwidth-bound: 32MB fp32 input @ 23.3 TB/s => ~1.4us floor. Compute is
// ~2 GFLOP after algebraic collapse => FP32 WMMA (v_wmma_f32_16x16x4_f32)
// keeps full precision with zero cost. B operands are pre-fragmentized into
// per-(step,lane) float2 tables so every WMMA B load is one coalesced b64.
// ---------------------------------------------------------------------------

typedef __attribute__((ext_vector_type(2))) float v2f;
typedef __attribute__((ext_vector_type(8))) float v8f;

#define NROWS 65536
#define FF    64
#define EE    16

// ws float-offset layout
#define WS_XCSUM   0        // [64]   column sums of Xc
#define WS_HSUM    64       // [8]    sum of h (pre-BN) per channel
#define WS_HSQ     72       // [8]    sum of h^2 per channel
#define WS_BSF     128      // [48*32 float2]  s-GEMM B fragments (p|xc|xa)
#define WS_BHF     3200     // [48*32 float2]  h-GEMM B fragments (cols 8..15 = 0)
#define WS_BQF     6272     // [96*32 float2]  Q-GEMM B fragments (6 channels)
#define WS_HPRE    12544    // [N*8]  pre-batchnorm h

// ---------------------------------------------------------------------------
// Kernel 1: zero reduction slots, build WMMA-fragment weight tables.
// Fragment layout (fp32 16x16x4 B operand): lane = khalf*16 + col,
// float2 = { B[k0][col], B[k0+1][col] }, k0 = s*4 + khalf*2 within channel.
// ---------------------------------------------------------------------------
__global__ __launch_bounds__(256) void setup_kernel(
    const float* __restrict__ W1, const float* __restrict__ B1,
    const float* __restrict__ W2, const float* __restrict__ B2,
    const float* __restrict__ lin1_w, float* __restrict__ ws)
{
    int t = threadIdx.x;
    if (t < 128) ws[t] = 0.0f;   // xc_sum + h stats (+pad)

    float2* BsF = (float2*)(ws + WS_BSF);
    float2* BhF = (float2*)(ws + WS_BHF);
    float2* BqF = (float2*)(ws + WS_BQF);

    // s/h tables: 48 steps = 3 channels (p, xc, xa) x 16 f-blocks
    for (int i = t; i < 48 * 32; i += 256) {
        int step = i >> 5, ln = i & 31;
        int c = step >> 4, s = step & 15;
        int col = ln & 15;
        int fb  = s * 4 + ((ln >> 4) << 1);
        float vs[2], vh[2];
        #pragma unroll
        for (int d = 0; d < 2; d++) {
            int f = fb + d;
            float src;
            if (c == 0)      src = W1[f * 16 + col] + W2[f * 16 + col];
            else if (c == 1) src = B1[f * 16 + col];
            else             src = B2[f * 16 + col];
            vs[d] = src;
            float acc = 0.0f;
            if (col < 8) {   // fold lin1_w over e; cols 8..15 stay zero
                const float* L = lin1_w + col * (FF * EE) + f * 16;
                for (int e = 0; e < 16; e++) {
                    float srcv = (c == 0) ? (W1[f * 16 + e] + W2[f * 16 + e])
                               : (c == 1) ? B1[f * 16 + e]
                                          : B2[f * 16 + e];
                    acc += srcv * L[e];
                }
            }
            vh[d] = acc;
        }
        BsF[i] = make_float2(vs[0], vs[1]);
        BhF[i] = make_float2(vh[0], vh[1]);
    }

    // Q table: 96 steps = 6 channels (p^2, xc^2, xa^2, p*xc, p*xa, p) x 16
    for (int i = t; i < 96 * 32; i += 256) {
        int step = i >> 5, ln = i & 31;
        int c = step >> 4, s = step & 15;
        int col = ln & 15;
        int fb  = s * 4 + ((ln >> 4) << 1);
        float vq[2];
        #pragma unroll
        for (int d = 0; d < 2; d++) {
            int f = fb + d;
            float w  = W1[f * 16 + col] + W2[f * 16 + col];
            float c1 = B1[f * 16 + col];
            float c2 = B2[f * 16 + col];
            float v;
            switch (c) {
                case 0:  v = w * w;         break;
                case 1:  v = c1 * c1;       break;
                case 2:  v = c2 * c2;       break;
                case 3:  v = 2.0f * w * c1; break;
                case 4:  v = 2.0f * w * c2; break;
                default: v = 2.0f * c1 * c2; break;  // channel p
            }
            vq[d] = v;
        }
        BqF[i] = make_float2(vq[0], vq[1]);
    }
}

// ---------------------------------------------------------------------------
// Kernel 2: column sums of Xc over N (for xc_mean).
// ---------------------------------------------------------------------------
__global__ __launch_bounds__(256) void colsum_kernel(
    const float* __restrict__ Xc, float* __restrict__ ws)
{
    __shared__ float red[256];
    int t  = threadIdx.x;
    int f  = t & 63;
    int rg = t >> 6;                       // 0..3
    int rbase = blockIdx.x * 256;
    float s = 0.0f;
    #pragma unroll 4
    for (int i = 0; i < 64; i++) {
        int r = rbase + rg + i * 4;
        s += Xc[(size_t)r * FF + f];
    }
    red[t] = s;
    __syncthreads();
    if (t < 64) {
        float tot = red[t] + red[t + 64] + red[t + 128] + red[t + 192];
        atomicAdd(&ws[WS_XCSUM + f], tot);
    }
}

// ---------------------------------------------------------------------------
// Kernel 3: main pass. 1 wave = one 16-row tile; 12 WMMAs per f-block step,
// fully unrolled (192 wmma/tile), no branches in the hot loop.
// ---------------------------------------------------------------------------
#define LROW 68   // padded LDS row stride (floats): conflict-free, 16B aligned

__global__ __launch_bounds__(128) void fm_main_kernel(
    const float* __restrict__ Xa, const float* __restrict__ Xc,
    const float* __restrict__ w1, const float* __restrict__ b1,
    const float* __restrict__ w2, const float* __restrict__ b2,
    const float* __restrict__ lin1_b,
    float* __restrict__ ws, float* __restrict__ out)
{
    __shared__ float lds[4 * (2 * 16 * LROW + 16)];
    const int wave = threadIdx.x >> 5;
    const int lane = threadIdx.x & 31;
    float* lxa  = &lds[wave * (2 * 16 * LROW + 16)];
    float* lxc  = lxa + 16 * LROW;
    float* lout = lxc + 16 * LROW;

    const int tile = blockIdx.x * 4 + wave;
    const int n0   = tile * 16;

    const float2* BsF = (const float2*)(ws + WS_BSF);
    const float2* BhF = (const float2*)(ws + WS_BHF);
    const float2* BqF = (const float2*)(ws + WS_BQF);
    __builtin_prefetch(BqF + lane, 0, 1);   // global_prefetch_b8

    // ---- stage 16x64 tiles of Xa/Xc into LDS (float4, coalesced) ----
    for (int i = lane; i < 16 * 16; i += 32) {
        int r = i >> 4, q = i & 15;
        float4 va = ((const float4*)(Xa + (size_t)(n0 + r) * FF))[q];
        float4 vc = ((const float4*)(Xc + (size_t)(n0 + r) * FF))[q];
        *(float4*)&lxa[r * LROW + q * 4] = va;
        *(float4*)&lxc[r * LROW + q * 4] = vc;
    }
    __syncthreads();

    // ---- WMMA fragment coordinates (ISA 7.12.2, fp32 16x16x4) ----
    const int mrow = lane & 15;           // A: M index
    const int ksub = (lane >> 4) * 2;     // A: K = base+ksub, base+ksub+1
    const float* rowa = lxa + mrow * LROW + ksub;
    const float* rowc = lxc + mrow * LROW + ksub;

    v8f acc_s = {}, acc_h = {}, acc_q = {};

#define LDB(tab, st)  do { float2 t2_ = (tab)[(st) * 32 + lane];              \
                           b.x = t2_.x; b.y = t2_.y; } while (0)
#define WM(accv)      accv = __builtin_amdgcn_wmma_f32_16x16x4_f32(           \
                          false, a, false, b, (short)0, accv, false, false)

    #pragma unroll
    for (int s = 0; s < 16; s++) {
        float xa0 = rowa[s * 4], xa1 = rowa[s * 4 + 1];   // ds_load_b64
        float xc0 = rowc[s * 4], xc1 = rowc[s * 4 + 1];   // ds_load_b64
        float p0 = xa0 * xc0, p1 = xa1 * xc1;
        v2f a, b;

        a.x = p0;        a.y = p1;          // channel p
        LDB(BsF, s);        WM(acc_s);
        LDB(BhF, s);        WM(acc_h);
        LDB(BqF, 80 + s);   WM(acc_q);      // Q cross term 2*B1*B2
        a.x = xc0;       a.y = xc1;         // channel xc
        LDB(BsF, 16 + s);   WM(acc_s);
        LDB(BhF, 16 + s);   WM(acc_h);
        a.x = xa0;       a.y = xa1;         // channel xa
        LDB(BsF, 32 + s);   WM(acc_s);
        LDB(BhF, 32 + s);   WM(acc_h);
        a.x = p0 * p0;   a.y = p1 * p1;     // Q: p^2
        LDB(BqF, s);        WM(acc_q);
        a.x = xc0 * xc0; a.y = xc1 * xc1;   // Q: xc^2
        LDB(BqF, 16 + s);   WM(acc_q);
        a.x = xa0 * xa0; a.y = xa1 * xa1;   // Q: xa^2
        LDB(BqF, 32 + s);   WM(acc_q);
        a.x = p0 * xc0;  a.y = p1 * xc1;    // Q: p*xc
        LDB(BqF, 48 + s);   WM(acc_q);
        a.x = p0 * xa0;  a.y = p1 * xa1;    // Q: p*xa
        LDB(BqF, 64 + s);   WM(acc_q);
    }
#undef LDB
#undef WM

    // ---- fm_second mean over e ----
    // D layout: lane l, vgpr v -> element [M = v + 8*(l>>4)][N = l&15]
    const int hlf = lane >> 4;
    float fm2[8];
    #pragma unroll
    for (int v = 0; v < 8; v++) {
        float t = 0.5f * (acc_s[v] * acc_s[v] - acc_q[v]);
        t += __shfl_xor(t, 1);
        t += __shfl_xor(t, 2);
        t += __shfl_xor(t, 4);
        t += __shfl_xor(t, 8);
        fm2[v] = t * (1.0f / 16.0f);
    }
    if ((lane & 15) == 0) {
        #pragma unroll
        for (int v = 0; v < 8; v++) lout[hlf * 8 + v] = fm2[v];
    }

    // ---- h (pre-batchnorm): store + batch stats ----
    float* hpre = ws + WS_HPRE;
    const int j = lane & 15;
    float hb = (j < 8) ? lin1_b[j] : 0.0f;
    float hsum = 0.0f, hsq = 0.0f;
    #pragma unroll
    for (int v = 0; v < 8; v++) {
        float hv = acc_h[v] + hb;          // cols 8..15 are exact zeros
        int r = v + 8 * hlf;
        if (j < 8) hpre[(size_t)(n0 + r) * 8 + j] = hv;
        hsum += hv;
        hsq  += hv * hv;
    }
    hsum += __shfl_xor(hsum, 16);
    hsq  += __shfl_xor(hsq, 16);
    if (lane < 8) {
        atomicAdd(&ws[WS_HSUM + lane], hsum);
        atomicAdd(&ws[WS_HSQ  + lane], hsq);
    }

    // ---- fm_first mean over f (uses xc_mean from colsum pass) ----
    const float invN = 1.0f / (float)NROWS;
    const int row   = lane & 15;
    const int fbase = hlf * 32;
    float acc1 = 0.0f;
    #pragma unroll 8
    for (int ff = 0; ff < 32; ff++) {
        int f = fbase + ff;
        float xa = lxa[row * LROW + f];
        float xc = lxc[row * LROW + f];
        float cst = w2[f] * (ws[WS_XCSUM + f] * invN) + b2[f];
        acc1 += (w1[f] * xa + b1[f]) * xc + xa * cst;
    }
    acc1 += __shfl_xor(acc1, 16);

    __syncthreads();   // publish lout within the block (uniform barrier)
    if (lane < 16) {
        out[n0 + lane] = acc1 * (1.0f / 64.0f) + lout[lane];
    }
}

// ---------------------------------------------------------------------------
// Kernel 4: batchnorm (population var) + tanh + lin2 + row mean; += into out.
// ---------------------------------------------------------------------------
__global__ __launch_bounds__(256) void finalize_kernel(
    const float* __restrict__ lin2_w, const float* __restrict__ lin2_b,
    const float* __restrict__ gamma, const float* __restrict__ beta,
    const float* __restrict__ ws, float* __restrict__ out)
{
    int n = blockIdx.x * blockDim.x + threadIdx.x;
    const float invN = 1.0f / (float)NROWS;
    const float* h = ws + WS_HPRE + (size_t)n * 8;

    float hn[8];
    #pragma unroll
    for (int j = 0; j < 8; j++) {
        float mu = ws[WS_HSUM + j] * invN;
        float vr = ws[WS_HSQ + j] * invN - mu * mu;
        float z  = (h[j] - mu) * rsqrtf(vr + 1e-5f) * gamma[j] + beta[j];
        hn[j] = tanhf(z);
    }
    float y = 0.0f;
    #pragma unroll
    for (int k = 0; k < 4; k++) {
        float acc = lin2_b[k];
        #pragma unroll
        for (int j = 0; j < 8; j++) acc += hn[j] * lin2_w[k * 8 + j];
        y += acc;
    }
    out[n] += y * 0.25f;
}

// ---------------------------------------------------------------------------
extern "C" void kernel_launch(void* const* d_in, const int* in_sizes, int n_in,
                              void* d_out, int out_size, void* d_ws, size_t ws_size,
                              hipStream_t stream)
{
    const float* Xa     = (const float*)d_in[0];
    const float* Xc     = (const float*)d_in[1];
    const float* w1     = (const float*)d_in[2];
    const float* b1     = (const float*)d_in[3];
    const float* w2     = (const float*)d_in[4];
    const float* b2     = (const float*)d_in[5];
    const float* W1     = (const float*)d_in[6];
    const float* B1     = (const float*)d_in[7];
    const float* W2     = (const float*)d_in[8];
    const float* B2     = (const float*)d_in[9];
    const float* lin1_w = (const float*)d_in[10];
    const float* lin1_b = (const float*)d_in[11];
    const float* gamma  = (const float*)d_in[12];
    const float* beta   = (const float*)d_in[13];
    const float* lin2_w = (const float*)d_in[14];
    const float* lin2_b = (const float*)d_in[15];
    float* ws  = (float*)d_ws;
    float* out = (float*)d_out;

    setup_kernel  <<<1,              256, 0, stream>>>(W1, B1, W2, B2, lin1_w, ws);
    colsum_kernel <<<NROWS / 256,    256, 0, stream>>>(Xc, ws);
    fm_main_kernel<<<NROWS / (16*4), 128, 0, stream>>>(Xa, Xc, w1, b1, w2, b2,
                                                       lin1_b, ws, out);
    finalize_kernel<<<NROWS / 256,   256, 0, stream>>>(lin2_w, lin2_b, gamma,
                                                       beta, ws, out);
}